// CRF_22548578304854
// MI455X (gfx1250) — compile-verified
//
#include <hip/hip_runtime.h>
#include <math.h>

typedef __attribute__((ext_vector_type(2))) float v2f;
typedef __attribute__((ext_vector_type(4))) float v4f;
typedef __attribute__((ext_vector_type(8))) float v8f;

#define NEGV (-10000.0f)
constexpr int Bn = 512, Sn = 1024, Tn = 64;
constexpr int LSTR = 68; // LDS row stride (floats): 16B-aligned rows, stride-4 mod
                         // 64 banks -> 16 rows hit 16 distinct banks (conflict-free)

// ---------------- score_y kernel: one block per batch ----------------
__global__ void __launch_bounds__(256) crf_score_kernel(
    const float* __restrict__ emission, const int* __restrict__ y,
    const float* __restrict__ mask, const float* __restrict__ transition,
    const int* __restrict__ start_p, const int* __restrict__ end_p,
    float* __restrict__ score) {
  __shared__ float red[256];
  const int b = blockIdx.x;
  const int t = threadIdx.x;
  const int start_idx = *start_p;
  const int end_idx = *end_p;
  float acc = 0.f;
  for (int s = t; s < Sn; s += 256) {
    int cur = y[b * Sn + s];
    int prev = (s == 0) ? start_idx : y[b * Sn + s - 1];
    float e = emission[((size_t)b * Sn + s) * Tn + cur];
    float tr = transition[prev * Tn + cur];
    acc += (e + tr) * mask[b * Sn + s];
  }
  red[t] = acc;
  __syncthreads();
  for (int off = 128; off > 0; off >>= 1) {
    if (t < off) red[t] += red[t + off];
    __syncthreads();
  }
  if (t == 0) score[b] = red[0] + transition[y[b * Sn + Sn - 1] * Tn + end_idx];
}

// ------------- forward-algorithm kernel: 16 batches per block -------------
// new_alpha[j] = M + log( sum_i exp(alpha_i - M) * expT[i][j] ) + emi[j]
// inner sum = (16xK=64) x (64x64) fp32 matmul -> V_WMMA_F32_16X16X4_F32 chain.
// The shift M need not be the exact row max: alpha spread per row is bounded
// (~±12), so we sample next step's shift from a freshly computed finite column
// (ping-ponged Mrow buffer; written only by the lane owning that column).
__global__ void __launch_bounds__(128) crf_forward_kernel(
    const float* __restrict__ emission, const float* __restrict__ mask,
    const float* __restrict__ transition,
    const int* __restrict__ start_p, const int* __restrict__ end_p,
    float* __restrict__ logz) {
  __shared__ __align__(16) float expTt[Tn * LSTR];   // [col][row] = exp(Tr[row][col])
  __shared__ __align__(16) float aS[16 * LSTR];      // exp(alpha - Mrow)
  __shared__ __align__(16) float alphaS[16 * LSTR];  // running log-alpha
  __shared__ __align__(16) float Mrow[2][16];        // ping-pong shift per row
  __shared__ __align__(16) float maskRow[16];

  const int t = threadIdx.x;
  const int lane = t & 31;
  const int wave = t >> 5;
  const int hi = lane >> 4;   // 0 or 1 (lane half)
  const int lr = lane & 15;
  const int n0 = wave * 16;   // this wave's 16 output columns
  const int b0 = blockIdx.x * 16;
  const int start_idx = *start_p;
  const int end_idx = *end_p;

  // init exp(transition) transposed, and initial alpha
  for (int i = t; i < Tn * Tn; i += 128) {
    int c = i >> 6, r = i & 63;
    expTt[c * LSTR + r] = __expf(transition[r * Tn + c]);
  }
  for (int i = t; i < 16 * Tn; i += 128) {
    int row = i >> 6, col = i & 63;
    alphaS[row * LSTR + col] = (col == start_idx) ? 0.0f : NEGV;
  }
  if (t < 16) Mrow[0][t] = 0.0f;  // max of initial alpha (start column == 0.0)
  __syncthreads();

  // Preload constant B fragments once: B is 4x16 f32 per k-step.
  // v0 = row(4kk + 2*hi), v1 = row(4kk + 2*hi + 1), col = n0 + lr.
  v2f bfrag[16];
#pragma unroll
  for (int kk = 0; kk < 16; ++kk) {
    int k = 4 * kk + 2 * hi;
    bfrag[kk] = *(const v2f*)&expTt[(n0 + lr) * LSTR + k];
  }

  // lane that publishes next step's shift: a column that is never the start
  // column (start column alpha -> -inf), owned by wave 0
  const int wcol = (start_idx == 1) ? 2 : 1;
  const bool writer = (wave == 0) && (lr == wcol);

  // phase-A mapping: each thread owns 8 consecutive columns of one row
  const int arow = t >> 3;
  const int acol = (t & 7) * 8;
  float* aIn  = &alphaS[arow * LSTR + acol];
  float* aOut = &aS[arow * LSTR + acol];

  const float* emiBase = emission + (size_t)b0 * Sn * Tn;

  for (int s = 0; s < Sn; ++s) {
    const int p = s & 1;
    // phase A: mask load + a = exp(alpha - M), vectorized b128 LDS traffic
    if (t >= 16 && t < 32) maskRow[t - 16] = mask[(size_t)(b0 + t - 16) * Sn + s];
    {
      const float Mv = Mrow[p][arow];
      v4f x0 = *(const v4f*)aIn;
      v4f x1 = *(const v4f*)(aIn + 4);
      v4f e0, e1;
#pragma unroll
      for (int j = 0; j < 4; ++j) {
        e0[j] = __expf(x0[j] - Mv);
        e1[j] = __expf(x1[j] - Mv);
      }
      *(v4f*)aOut = e0;
      *(v4f*)(aOut + 4) = e1;
    }
    __syncthreads();

    // phase B: 16-deep WMMA fp32 chain, then log + emission, masked update
    v8f c = {};
#pragma unroll
    for (int kk = 0; kk < 16; ++kk) {
      // A 16x4 f32: lanes 0-15 row=lr, k = 4kk+{0,1}; lanes 16-31 k = 4kk+{2,3}
      v2f a = *(const v2f*)&aS[lr * LSTR + 4 * kk + 2 * hi];
      c = __builtin_amdgcn_wmma_f32_16x16x4_f32(false, a, false, bfrag[kk],
                                                (short)0, c, false, false);
    }
    float Mr[8], mk[8];
#pragma unroll
    for (int r = 0; r < 8; ++r) {
      Mr[r] = Mrow[p][r + 8 * hi];
      mk[r] = maskRow[r + 8 * hi];
    }
    const int col = n0 + lr;
#pragma unroll
    for (int r = 0; r < 8; ++r) {
      int row = r + 8 * hi;  // C/D layout: vgpr r -> M=r (lo half) / M=r+8 (hi half)
      const float* ep = emiBase + (size_t)row * Sn * Tn + (size_t)s * Tn + col;
      // c is never denormal (>= ~0.3, or exactly 0 -> -inf): raw v_log_f32 ok
      float na = 0.6931471805599453f * __builtin_amdgcn_logf(c[r]) + Mr[r] + *ep;
      if (s + 1 < Sn) __builtin_prefetch(ep + Tn, 0, 3);  // next step's emission
      if (mk[r] > 0.0f) alphaS[row * LSTR + col] = na;
      if (writer) Mrow[1 - p][row] = (mk[r] > 0.0f) ? na : Mr[r];
    }
    __syncthreads();
  }

  // final: log_z[b] = logsumexp_j(alpha[b][j] + Tr[j][end])  (exact max here)
  if (t < 16) {
    float m = -INFINITY;
    for (int j = 0; j < Tn; ++j)
      m = fmaxf(m, alphaS[t * LSTR + j] + transition[j * Tn + end_idx]);
    float sum = 0.f;
    for (int j = 0; j < Tn; ++j)
      sum += __expf(alphaS[t * LSTR + j] + transition[j * Tn + end_idx] - m);
    logz[b0 + t] = m + __logf(sum);
  }
}

// ---------------- final reduction: mean(logz - score) ----------------
__global__ void __launch_bounds__(256) crf_reduce_kernel(
    const float* __restrict__ score, const float* __restrict__ logz,
    float* __restrict__ out) {
  __shared__ float red[256];
  const int t = threadIdx.x;
  float acc = 0.f;
  for (int b = t; b < Bn; b += 256) acc += logz[b] - score[b];
  red[t] = acc;
  __syncthreads();
  for (int off = 128; off > 0; off >>= 1) {
    if (t < off) red[t] += red[t + off];
    __syncthreads();
  }
  if (t == 0) out[0] = red[0] / (float)Bn;
}

extern "C" void kernel_launch(void* const* d_in, const int* in_sizes, int n_in,
                              void* d_out, int out_size, void* d_ws, size_t ws_size,
                              hipStream_t stream) {
  (void)in_sizes; (void)n_in; (void)out_size; (void)ws_size;
  const float* emission   = (const float*)d_in[0];
  const int*   y          = (const int*)d_in[1];
  const float* mask       = (const float*)d_in[2];
  const float* transition = (const float*)d_in[3];
  const int*   start_p    = (const int*)d_in[4];
  const int*   end_p      = (const int*)d_in[5];

  float* wsf   = (float*)d_ws;
  float* score = wsf;        // [512]
  float* logz  = wsf + Bn;   // [512]
  float* out   = (float*)d_out;

  crf_score_kernel<<<Bn, 256, 0, stream>>>(emission, y, mask, transition,
                                           start_p, end_p, score);
  crf_forward_kernel<<<Bn / 16, 128, 0, stream>>>(emission, mask, transition,
                                                  start_p, end_p, logz);
  crf_reduce_kernel<<<1, 256, 0, stream>>>(score, logz, out);
}